// KNNGraph_90142773609276
// MI455X (gfx1250) — compile-verified
//
#include <hip/hip_runtime.h>

typedef float v2f __attribute__((ext_vector_type(2)));
typedef float v8f __attribute__((ext_vector_type(8)));

#define NPTS  4096
#define BATCH 8
#define CHAN  64
#define KNBR  20
#define TILES (NPTS / 16)

// f64 compare-exchange without llvm.canonicalize overhead: keys are finite
// positive doubles (never NaN), so v_min/max_num are safe directly.
__device__ __forceinline__ void cmpx_f64(double& a, double& b)
{
    double lo, hi;
    asm("v_min_num_f64 %0, %1, %2" : "=v"(lo) : "v"(a), "v"(b));
    asm("v_max_num_f64 %0, %1, %2" : "=v"(hi) : "v"(a), "v"(b));
    a = lo;
    b = hi;
}

// ---------------------------------------------------------------------------
// Kernel 1: fused pairwise-distance (V_WMMA_F32_16X16X4_F32) + exact top-20.
// Block = 256 threads (8 waves); each wave owns 16 queries.
// LDS A-operand (per candidate i):
//   pairs[i]        = ( x_i,  y_i )      (lanes 0-15 : K=0,1)
//   pairs[4096 + i] = ( z_i, |p_i|^2 )   (lanes 16-31: K=2,3)
// B-operand per lane (query q): half0 = (-2qx,-2qy), half1 = (-2qz, 1)
//   => D(M,N) = |p_M|^2 - 2 p_M . q_N ;  dist = |q|^2 + D
// Selection keys: double(dist) | idx packed into the 29 zero low-mantissa
// bits -> exact dist order, idx tie-break, sorted with v_min/max_num_f64.
// ---------------------------------------------------------------------------
__global__ void __launch_bounds__(256)
knn_topk_kernel(const float* __restrict__ coords, int* __restrict__ knn)
{
    extern __shared__ float smem[];
    v2f* pairs = (v2f*)smem;                 // 2*4096 v2f = 64 KB

    const int tid  = threadIdx.x;
    const int b    = blockIdx.x >> 5;        // 32 query-blocks per batch
    const int qblk = blockIdx.x & 31;
    const float* cb = coords + b * 3 * NPTS; // (3, N) for this batch

    for (int i = tid; i < NPTS; i += 256) {
        float x = cb[i], y = cb[NPTS + i], z = cb[2 * NPTS + i];
        pairs[i]        = (v2f){x, y};
        pairs[NPTS + i] = (v2f){z, x * x + y * y + z * z};
    }
    __syncthreads();

    const int lane = tid & 31;
    const int w    = tid >> 5;
    const int half = lane >> 4;
    const int m16  = lane & 15;
    const int q    = qblk * 128 + w * 16 + m16;

    float qx = cb[q], qy = cb[NPTS + q], qz = cb[2 * NPTS + q];
    float qn = qx * qx + qy * qy + qz * qz;
    v2f bmat = half ? (v2f){-2.f * qz, 1.f} : (v2f){-2.f * qx, -2.f * qy};

    double key[KNBR];
#pragma unroll
    for (int k = 0; k < KNBR; ++k) key[k] = 1e300;

    const int abase = half * NPTS + m16;
    v2f a = pairs[abase];                    // prefetch tile 0

    for (int t = 0; t < TILES; ++t) {
        // prefetch next tile's A fragment (wraps harmlessly at t=255)
        v2f an = pairs[abase + (((t + 1) & (TILES - 1)) << 4)];

        v8f acc = {0.f, 0.f, 0.f, 0.f, 0.f, 0.f, 0.f, 0.f};
        v8f d = __builtin_amdgcn_wmma_f32_16x16x4_f32(
            /*neg_a=*/false, a, /*neg_b=*/false, bmat,
            /*c_mod=*/(short)0, acc, /*reuse_a=*/false, /*reuse_b=*/false);

        const int c0 = t * 16 + half * 8;    // this lane's first candidate
#pragma unroll
        for (int r = 0; r < 8; ++r) {
            float dist = fmaxf(qn + d[r], 1e-12f);
            int   cand = c0 + r;
            // exact (dist, idx) key: f32->f64 leaves 29 zero mantissa LSBs
            double kk = __longlong_as_double(
                __double_as_longlong((double)dist) | (long long)cand);
            if (cand != q && kk < key[KNBR - 1]) {
                key[KNBR - 1] = kk;
#pragma unroll
                for (int s = KNBR - 1; s > 0; --s)
                    cmpx_f64(key[s - 1], key[s]);   // 2 VALU per level
            }
        }
        a = an;
    }

    // merge lane l (cands 0-7 mod 16) with lane l+16 (cands 8-15 mod 16)
    __syncthreads();                          // done reading pairs; reuse LDS
    double* mb = (double*)smem;               // 8*32*20*8 B = 40 KB
    const int sbase = (w * 32 + lane) * KNBR;
#pragma unroll
    for (int k = 0; k < KNBR; ++k) mb[sbase + k] = key[k];
    __syncthreads();

    if (lane < 16) {
        const int baseB = (w * 32 + lane + 16) * KNBR;
        int* outp = knn + (b * NPTS + q) * KNBR;
        int i = 0, j = 0;
        for (int k = 0; k < KNBR; ++k) {
            double da = mb[sbase + i];
            double db = mb[baseB + j];
            bool takeA = da < db;             // keys unique: no tie handling
            double sel = takeA ? da : db;
            outp[k] = (int)(__double_as_longlong(sel) & 0xFFF);
            if (takeA) ++i; else ++j;
        }
    }
}

// ---------------------------------------------------------------------------
// Kernel 2: gather + concat. out[b, c, n, k]:
//   c <  64 : feats[b,c,n]
//   c >= 64 : feats[b,c-64, knn[b,n,k]] - feats[b,c-64,n]
// One thread writes 4 consecutive k's (float4, 16 B aligned: row pitch 80 B).
// feats (8 MB) stays L2-resident; the 335 MB write stream is coalesced and
// sets the ~14 us roofline for the whole op at 23.3 TB/s.
// ---------------------------------------------------------------------------
__global__ void __launch_bounds__(256)
knn_gather_kernel(const float* __restrict__ feats,
                  const int* __restrict__ knn,
                  float* __restrict__ out)
{
    int tid = blockIdx.x * 256 + threadIdx.x;    // total B*128*N*5
    int g = tid % 5;
    int r = tid / 5;
    int n = r & (NPTS - 1); r >>= 12;
    int c = r & 127;
    int b = r >> 7;

    const float* fb = feats + (size_t)b * CHAN * NPTS;
    float4 v;
    if (c < CHAN) {
        float fc = fb[c * NPTS + n];
        v.x = fc; v.y = fc; v.z = fc; v.w = fc;
    } else {
        const float* fr = fb + (c - CHAN) * NPTS;
        float fc = fr[n];
        int4 id = *(const int4*)(knn + ((b * NPTS + n) * KNBR + g * 4));
        v.x = fr[id.x] - fc;
        v.y = fr[id.y] - fc;
        v.z = fr[id.z] - fc;
        v.w = fr[id.w] - fc;
    }
    *(float4*)(out + (size_t)((b * 128 + c) * NPTS + n) * KNBR + g * 4) = v;
}

// ---------------------------------------------------------------------------
extern "C" void kernel_launch(void* const* d_in, const int* in_sizes, int n_in,
                              void* d_out, int out_size, void* d_ws, size_t ws_size,
                              hipStream_t stream)
{
    const float* coords = (const float*)d_in[0];   // (8, 3, 4096)
    const float* feats  = (const float*)d_in[1];   // (8, 64, 4096, 1)
    float* out = (float*)d_out;                    // (8, 128, 4096, 20)
    int*   knn = (int*)d_ws;                       // needs 8*4096*20*4 = 2.62 MB

    const size_t shmem = 2 * NPTS * sizeof(v2f);   // 64 KB
    knn_topk_kernel<<<dim3(BATCH * 32), dim3(256), shmem, stream>>>(coords, knn);

    const int total_threads = BATCH * 128 * NPTS * 5;   // 20,971,520
    knn_gather_kernel<<<dim3(total_threads / 256), dim3(256), 0, stream>>>(feats, knn, out);
}